// CrossGTPNet_17463337025772
// MI455X (gfx1250) — compile-verified
//
#include <hip/hip_runtime.h>
#include <math.h>

// BatchGATConv forward for MI455X (gfx1250).
//   N=50000, B=4, DIN=256, H=4, DOUT=16, E=800000, neg_slope=0.2
// Kernel 1: fused f32 WMMA GEMM (ft = feat@W_fc^T, res = feat@W_res^T -> d_out)
//           weights staged into LDS by the Tensor Data Mover (padded rows ->
//           conflict-free ds_load b64 B fragments); 32-row M tile per wave so
//           8 LDS loads feed 16 WMMAs per K step; A streams double-buffered.
// Kernels 2-5: edge softmax (native f32 atomic max / add) + weighted scatter.
// Workspace layout (floats): ft[M*64] | el[M*4] | er[M*4] | emax[M*4] | denom[M*4]

#define DIN_   256
#define B_     4
#define H_     4
#define DOUT_  16
#define NSLOPE 0.2f

// LDS row stride for weights: 256 data DWORDs + 4 pad DWORDs (TDM pad feature)
#define WROW_  260
#define WMAT_  (64 * WROW_)              // floats per staged matrix (16640)
#define LDSBYTES_ (2 * WMAT_ * 4)        // 133120 bytes dynamic LDS

typedef __attribute__((ext_vector_type(2))) float        v2f;
typedef __attribute__((ext_vector_type(8))) float        v8f;
typedef __attribute__((ext_vector_type(4))) unsigned int v4u;
typedef __attribute__((ext_vector_type(4))) int          v4i;
typedef __attribute__((ext_vector_type(8))) int          v8i;

__device__ __forceinline__ float leaky(float v) {
    return v > 0.0f ? v : NSLOPE * v;
}

// ---------------------------------------------------------------------------
// TDM: DMA one 64x256 f32 weight matrix from global into LDS at byte offset
// lds_off, inserting 4 pad DWORDs after every 256 DWORDs (row stride 260).
// Descriptor encoding per CDNA5 ISA (2-D tensor,
// tile == tensor, groups 2/3 zero). 6-arg builtin on this toolchain.
// ---------------------------------------------------------------------------
__device__ __forceinline__ void tdm_load_weight(const float* W, unsigned lds_off) {
    unsigned long long ga = (unsigned long long)(uintptr_t)W;

    v4u g0;
    g0.x = 1u;                                        // count=1, user desc
    g0.y = lds_off;                                   // lds_addr (bytes)
    g0.z = (unsigned)(ga & 0xFFFFFFFFu);              // global_addr[31:0]
    g0.w = (unsigned)((ga >> 32) & 0x01FFFFFFu)       // global_addr[56:32]
         | (2u << 30);                                // type = 2 ("image")

    v8i g1;
    g1[0] = (int)((2u << 16)                          // data_size = 4 bytes
                | (1u << 20)                          // pad_enable
                | (7u << 22)                          // pad_interval: 256 DW
                | (3u << 25));                        // pad_amount: 4 DW
    g1[1] = (int)(256u << 16);                        // tensor_dim0 = 256
    g1[2] = (int)(64u  << 16);                        // tensor_dim1 = 64
    g1[3] = (int)(256u << 16);                        // tile_dim0  = 256
    g1[4] = (int)(64u);                               // tile_dim1  = 64
    g1[5] = (int)(256u);                              // tensor_dim0_stride=256
    g1[6] = 0;
    g1[7] = 0;

    v4i gz4 = {0, 0, 0, 0};
    v8i gz8 = {0, 0, 0, 0, 0, 0, 0, 0};
    __builtin_amdgcn_tensor_load_to_lds(g0, g1, gz4, gz4, gz8, 0);
}

// ---------------------------------------------------------------------------
// Kernel 1: per-wave 32-row M tile (two 16-row sub-tiles sharing B fragments).
//   acc_fc[s][t] / acc_res[s][t]: sub-tile s, head-column tile t.
// C layout: VGPR j -> row rbase + j + 8*(lane>>4), col t*16 + (lane&15).
// Requires M % 32 == 0 (holds: M = N*B = 200000).
// ---------------------------------------------------------------------------
__global__ __launch_bounds__(256) void gat_gemm_wmma(
    const float* __restrict__ feat,
    const float* __restrict__ W_fc,
    const float* __restrict__ W_res,
    const float* __restrict__ attn_l,
    const float* __restrict__ attn_r,
    float* __restrict__ ft_out,     // (M,64)
    float* __restrict__ el_out,     // (M,4)
    float* __restrict__ er_out,     // (M,4)
    float* __restrict__ out,        // (M,64) <- residual written here
    int M)
{
    extern __shared__ float lds[];  // [2][64][260]

    const int lane = threadIdx.x & 31;
    const int wave = threadIdx.x >> 5;
    const int tile = blockIdx.x * 8 + wave;
    const int r0   = tile * 32;

    // Stage W_fc and W_res into LDS via the Tensor Data Mover (wave 0 only;
    // scalar branch keeps other waves from issuing duplicate DMAs).
    if (threadIdx.x < 32) {
        tdm_load_weight(W_fc, 0u);
        tdm_load_weight(W_res, (unsigned)(WMAT_ * 4));
        __builtin_amdgcn_s_wait_tensorcnt(0);
    }
    __syncthreads();

    if (r0 >= M) return;                       // wave-uniform: EXEC all-1s

    const int mlo   = lane & 15;               // row-in-subtile / col-in-tile
    const int half  = lane >> 4;
    const int khalf = half * 2;                // K sub-offset {0,2}

    const float* arow0 = feat + (size_t)(r0 + mlo) * DIN_ + khalf;
    const float* arow1 = arow0 + (size_t)16 * DIN_;
    __builtin_prefetch(arow0, 0, 0);           // global_prefetch_b8
    __builtin_prefetch(arow1, 0, 0);

    v8f acc_fc[2][4]  = {};
    v8f acc_res[2][4] = {};

    // Per-lane LDS base pointers for the 8 B-fragment streams.
    const float* bfc[4];
    const float* brs[4];
    #pragma unroll
    for (int t = 0; t < 4; ++t) {
        bfc[t] = lds + (size_t)(t * 16 + mlo) * WROW_ + khalf;
        brs[t] = bfc[t] + WMAT_;
    }

    // K loop: 8 LDS B loads feed 16 WMMAs; both A streams double-buffered.
    v2f a0, a1;
    a0.x = arow0[0]; a0.y = arow0[1];
    a1.x = arow1[0]; a1.y = arow1[1];
    for (int k = 0; k < DIN_; k += 4) {
        v2f a0n = a0, a1n = a1;
        const int kn = k + 4;
        if (kn < DIN_) {                       // uniform scalar branch
            a0n.x = arow0[kn]; a0n.y = arow0[kn + 1];
            a1n.x = arow1[kn]; a1n.y = arow1[kn + 1];
        }
        #pragma unroll
        for (int t = 0; t < 4; ++t) {
            v2f bf, br;
            bf.x = bfc[t][k];
            bf.y = bfc[t][k + 1];
            br.x = brs[t][k];
            br.y = brs[t][k + 1];
            acc_fc[0][t] = __builtin_amdgcn_wmma_f32_16x16x4_f32(
                false, a0, false, bf, (short)0, acc_fc[0][t], false, false);
            acc_fc[1][t] = __builtin_amdgcn_wmma_f32_16x16x4_f32(
                false, a1, false, bf, (short)0, acc_fc[1][t], false, false);
            acc_res[0][t] = __builtin_amdgcn_wmma_f32_16x16x4_f32(
                false, a0, false, br, (short)0, acc_res[0][t], false, false);
            acc_res[1][t] = __builtin_amdgcn_wmma_f32_16x16x4_f32(
                false, a1, false, br, (short)0, acc_res[1][t], false, false);
        }
        a0 = a0n;
        a1 = a1n;
    }

    // Epilogue per 16-row sub-tile: store ft and res, then reduce el/er.
    #pragma unroll
    for (int s = 0; s < 2; ++s) {
        const int rbase = r0 + s * 16;

        #pragma unroll
        for (int t = 0; t < 4; ++t) {
            #pragma unroll
            for (int j = 0; j < 8; ++j) {
                const size_t row = (size_t)(rbase + j + 8 * half);
                const size_t idx = row * 64 + t * 16 + mlo;
                ft_out[idx] = acc_fc[s][t][j];
                out[idx]    = acc_res[s][t][j];
            }
        }

        // el/er: 16-wide dot of ft row-fragment against attn vectors, reduced
        // across each 16-lane half with xor shuffles (wave32).
        #pragma unroll
        for (int t = 0; t < 4; ++t) {
            const float al = attn_l[t * DOUT_ + mlo];
            const float ar = attn_r[t * DOUT_ + mlo];
            #pragma unroll
            for (int j = 0; j < 8; ++j) {
                float pl = acc_fc[s][t][j] * al;
                float pr = acc_fc[s][t][j] * ar;
                #pragma unroll
                for (int m = 1; m <= 8; m <<= 1) {
                    pl += __shfl_xor(pl, m, 32);
                    pr += __shfl_xor(pr, m, 32);
                }
                if (mlo == t) {
                    const size_t row = (size_t)(rbase + j + 8 * half);
                    el_out[row * H_ + t] = pl;
                    er_out[row * H_ + t] = pr;
                }
            }
        }
    }
}

// ---------------------------------------------------------------------------
// Kernel 2: init emax = -inf, denom = 0  (M*4 entries each)
// ---------------------------------------------------------------------------
__global__ __launch_bounds__(256) void gat_init(
    float* __restrict__ emax, float* __restrict__ denom, int n)
{
    int i = blockIdx.x * 256 + threadIdx.x;
    if (i < n) {
        emax[i]  = -INFINITY;
        denom[i] = 0.0f;
    }
}

// ---------------------------------------------------------------------------
// Kernel 3: per-edge atomic max of leaky score into emax[dst].
// el/er are (node, B*H=16) contiguous -> float4 gathers.
// ---------------------------------------------------------------------------
__global__ __launch_bounds__(256) void gat_edge_max(
    const int* __restrict__ esrc, const int* __restrict__ edst,
    const float* __restrict__ el, const float* __restrict__ er,
    float* __restrict__ emax, int E)
{
    int e = blockIdx.x * 256 + threadIdx.x;
    if (e >= E) return;
    const size_t s = (size_t)esrc[e] * 16;
    const size_t d = (size_t)edst[e] * 16;
    const float4* lp = (const float4*)(el + s);
    const float4* rp = (const float4*)(er + d);
    #pragma unroll
    for (int q = 0; q < 4; ++q) {
        float4 a = lp[q], b = rp[q];
        float v0 = leaky(a.x + b.x), v1 = leaky(a.y + b.y);
        float v2 = leaky(a.z + b.z), v3 = leaky(a.w + b.w);
        __hip_atomic_fetch_max(&emax[d + q * 4 + 0], v0, __ATOMIC_RELAXED, __HIP_MEMORY_SCOPE_AGENT);
        __hip_atomic_fetch_max(&emax[d + q * 4 + 1], v1, __ATOMIC_RELAXED, __HIP_MEMORY_SCOPE_AGENT);
        __hip_atomic_fetch_max(&emax[d + q * 4 + 2], v2, __ATOMIC_RELAXED, __HIP_MEMORY_SCOPE_AGENT);
        __hip_atomic_fetch_max(&emax[d + q * 4 + 3], v3, __ATOMIC_RELAXED, __HIP_MEMORY_SCOPE_AGENT);
    }
}

// ---------------------------------------------------------------------------
// Kernel 4: per-edge exp(e - emax[dst]) accumulated into denom[dst].
// ---------------------------------------------------------------------------
__global__ __launch_bounds__(256) void gat_edge_sum(
    const int* __restrict__ esrc, const int* __restrict__ edst,
    const float* __restrict__ el, const float* __restrict__ er,
    const float* __restrict__ emax, float* __restrict__ denom, int E)
{
    int e = blockIdx.x * 256 + threadIdx.x;
    if (e >= E) return;
    const size_t s = (size_t)esrc[e] * 16;
    const size_t d = (size_t)edst[e] * 16;
    const float4* lp = (const float4*)(el + s);
    const float4* rp = (const float4*)(er + d);
    const float4* mp = (const float4*)(emax + d);
    #pragma unroll
    for (int q = 0; q < 4; ++q) {
        float4 a = lp[q], b = rp[q], m = mp[q];
        float e0 = __expf(leaky(a.x + b.x) - m.x);
        float e1 = __expf(leaky(a.y + b.y) - m.y);
        float e2 = __expf(leaky(a.z + b.z) - m.z);
        float e3 = __expf(leaky(a.w + b.w) - m.w);
        __hip_atomic_fetch_add(&denom[d + q * 4 + 0], e0, __ATOMIC_RELAXED, __HIP_MEMORY_SCOPE_AGENT);
        __hip_atomic_fetch_add(&denom[d + q * 4 + 1], e1, __ATOMIC_RELAXED, __HIP_MEMORY_SCOPE_AGENT);
        __hip_atomic_fetch_add(&denom[d + q * 4 + 2], e2, __ATOMIC_RELAXED, __HIP_MEMORY_SCOPE_AGENT);
        __hip_atomic_fetch_add(&denom[d + q * 4 + 3], e3, __ATOMIC_RELAXED, __HIP_MEMORY_SCOPE_AGENT);
    }
}

// ---------------------------------------------------------------------------
// Kernel 5: one thread per (edge, channel i = b*4+h): recompute a = softmax
// coefficient, scatter ft[src]*a into out[dst] (16 f32 atomics, L2-resident).
// ---------------------------------------------------------------------------
__global__ __launch_bounds__(256) void gat_edge_scatter(
    const int* __restrict__ esrc, const int* __restrict__ edst,
    const float* __restrict__ el, const float* __restrict__ er,
    const float* __restrict__ emax, const float* __restrict__ denom,
    const float* __restrict__ ft, float* __restrict__ out, int E)
{
    long long t = (long long)blockIdx.x * 256 + threadIdx.x;
    if (t >= (long long)E * 16) return;
    const int e = (int)(t >> 4);
    const int i = (int)(t & 15);
    const size_t s = (size_t)esrc[e] * 16 + i;
    const size_t d = (size_t)edst[e] * 16 + i;
    float v = leaky(el[s] + er[d]);
    float a = __expf(v - emax[d]) / denom[d];
    const float4* fp = (const float4*)(ft + s * 16);
    float* op = out + d * 16;
    #pragma unroll
    for (int q = 0; q < 4; ++q) {
        float4 f = fp[q];
        __hip_atomic_fetch_add(op + q * 4 + 0, f.x * a, __ATOMIC_RELAXED, __HIP_MEMORY_SCOPE_AGENT);
        __hip_atomic_fetch_add(op + q * 4 + 1, f.y * a, __ATOMIC_RELAXED, __HIP_MEMORY_SCOPE_AGENT);
        __hip_atomic_fetch_add(op + q * 4 + 2, f.z * a, __ATOMIC_RELAXED, __HIP_MEMORY_SCOPE_AGENT);
        __hip_atomic_fetch_add(op + q * 4 + 3, f.w * a, __ATOMIC_RELAXED, __HIP_MEMORY_SCOPE_AGENT);
    }
}

// ---------------------------------------------------------------------------
extern "C" void kernel_launch(void* const* d_in, const int* in_sizes, int n_in,
                              void* d_out, int out_size, void* d_ws, size_t ws_size,
                              hipStream_t stream)
{
    const float* feat   = (const float*)d_in[0];
    const float* W_fc   = (const float*)d_in[1];
    const float* attn_l = (const float*)d_in[2];
    const float* attn_r = (const float*)d_in[3];
    const float* W_res  = (const float*)d_in[4];
    const int*   esrc   = (const int*)d_in[5];
    const int*   edst   = (const int*)d_in[6];

    const int N = in_sizes[0] / (B_ * DIN_);
    const int M = N * B_;                  // 200000 rows of the flattened GEMM
    const int E = in_sizes[5];

    float* ft    = (float*)d_ws;           // M*64
    float* el    = ft    + (size_t)M * 64; // M*4
    float* er    = el    + (size_t)M * 4;
    float* emax  = er    + (size_t)M * 4;
    float* denom = emax  + (size_t)M * 4;
    float* out   = (float*)d_out;

    const int tiles  = (M + 31) / 32;      // 32-row tiles
    const int gblk   = (tiles + 7) / 8;    // 8 waves per block
    gat_gemm_wmma<<<gblk, 256, LDSBYTES_, stream>>>(
        feat, W_fc, W_res, attn_l, attn_r, ft, el, er, out, M);

    const int segN = M * 4;
    gat_init<<<(segN + 255) / 256, 256, 0, stream>>>(emax, denom, segN);

    gat_edge_max<<<(E + 255) / 256, 256, 0, stream>>>(esrc, edst, el, er, emax, E);
    gat_edge_sum<<<(E + 255) / 256, 256, 0, stream>>>(esrc, edst, el, er, emax, denom, E);

    const long long sc = (long long)E * 16;
    gat_edge_scatter<<<(int)((sc + 255) / 256), 256, 0, stream>>>(
        esrc, edst, el, er, emax, denom, ft, out, E);
}